// RelativePosition_66520453481102
// MI455X (gfx1250) — compile-verified
//
#include <hip/hip_runtime.h>
#include <hip/hip_bf16.h>
#include <stdint.h>

// Problem constants from the reference
#define RP_BINS      32
#define RP_CZ        128               // channels per pair
#define RP_NCLS      66                // 2*BINS + 2 classes
#define RP_L         768
#define RP_ZERO_ROW  66                // extra all-zero row for masked pairs
#define RP_ROWS      67
#define RP_ROW_BYTES (RP_CZ * 4)       // 512 B per class row
#define RP_W_ELEMS   (RP_NCLS * RP_CZ) // 8448 floats = 33,792 B

typedef unsigned int v4u __attribute__((ext_vector_type(4)));
typedef int          v8i __attribute__((ext_vector_type(8)));
typedef int          v4i __attribute__((ext_vector_type(4)));

#ifndef RP_HAS_TDM
#if defined(__has_builtin)
#if __has_builtin(__builtin_amdgcn_tensor_load_to_lds)
#define RP_HAS_TDM 1
#endif
#endif
#endif
#ifndef RP_HAS_TDM
#define RP_HAS_TDM 0
#endif

// One block per (b, i) output slab: writes L * CZ floats = 384 KB.
// Phase 1: wave0 TDM-DMAs W (66x128 f32, contiguous) into LDS while all
//          waves compute per-j class offsets + zero the masked-pair row;
//          then bias is fused in place: T[c][k] = W[c][k] + bias[k].
// Phase 2: each wave streams one 512B output row per iteration via
//          CDNA5 async LDS->global NT stores (no VGPR staging).
__global__ __launch_bounds__(256) void relpos_async_kernel(
    const int* __restrict__ residue_index,
    const unsigned char* __restrict__ residue_mask,   // jnp bool = 1 byte
    const float* __restrict__ W,                      // [66, 128]
    const float* __restrict__ bias,                   // [128]
    float* __restrict__ out,                          // [B, L, L, 128]
    int Lsz)
{
    __shared__ float    T[RP_ROWS * RP_CZ];           // 34,304 B
    __shared__ unsigned offs[RP_L];                   //  3,072 B

    const int t = threadIdx.x;
    // LDS byte address = low 32 bits of the flat shared pointer (aperture rule).
    const unsigned tbase = (unsigned)(uint64_t)(uintptr_t)&T[0];

    // ---- Phase 1a: bulk-load W -> T via the Tensor Data Mover (wave 0 only).
#if RP_HAS_TDM
    if (t < 32) {
        const unsigned long long ga = (unsigned long long)(uintptr_t)W;
        // D# group 0: count=1 | lds_addr | global_addr[56:0] | type=2
        v4u g0;
        g0.x = 0x1u;                                          // count=1, user desc
        g0.y = tbase;                                         // lds_addr (bytes)
        g0.z = (unsigned)(ga & 0xFFFFFFFFu);                  // global_addr[31:0]
        g0.w = (unsigned)((ga >> 32) & 0x01FFFFFFu)           // global_addr[56:32]
             | 0x80000000u;                                   // type=2 ("image")
        // D# group 1: data_size=4B; 1D tensor of 8448 elems as a 1-line 2D tile
        const unsigned NE = (unsigned)RP_W_ELEMS;             // 8448 (fits tile_dim0 u16)
        v8i g1;
        g1.s0 = (int)(0x2u << 16);                            // wg_mask=0, data_size=2 (4B)
        g1.s1 = (int)((NE & 0xFFFFu) << 16);                  // tensor_dim0[15:0]
        g1.s2 = (int)(((NE >> 16) & 0xFFFFu) | (1u << 16));   // tensor_dim0[31:16] | tensor_dim1=1
        g1.s3 = (int)((NE & 0xFFFFu) << 16);                  // tensor_dim1[31:16]=0 | tile_dim0=NE
        g1.s4 = 1;                                            // tile_dim1=1, tile_dim2=0
        g1.s5 = (int)NE;                                      // tensor_dim0_stride[31:0]
        g1.s6 = (int)((NE & 0xFFFFu) << 16);                  // stride0[47:32]=0 | stride1[15:0]
        g1.s7 = 0;                                            // stride1[47:16]=0
        v4i z4 = {0, 0, 0, 0};
#if defined(__clang_major__) && (__clang_major__ >= 23)
        v8i z8 = {0, 0, 0, 0, 0, 0, 0, 0};
        __builtin_amdgcn_tensor_load_to_lds(g0, g1, z4, z4, z8, 0);
#else
        __builtin_amdgcn_tensor_load_to_lds(g0, g1, z4, z4, 0);
#endif
    }
#else
    for (int e = t; e < RP_W_ELEMS; e += 256) T[e] = W[e];
#endif

    // ---- Phase 1b (overlaps TDM): zero row 66; per-j class -> row byte offset
    for (int k = t; k < RP_CZ; k += 256) T[RP_ZERO_ROW * RP_CZ + k] = 0.0f;

    const int bi   = (int)blockIdx.x;                 // = b*L + i
    const int b    = bi / Lsz;
    const int i    = bi - b * Lsz;
    const int ri_i = residue_index[b * Lsz + i];
    const bool m_i = residue_mask[b * Lsz + i] != 0;

    for (int j = t; j < Lsz; j += 256) {
        int d = residue_index[b * Lsz + j] - ri_i;    // diff[b,i,j] = ri[j]-ri[i]
        d = d < -RP_BINS ? -RP_BINS : (d > RP_BINS ? RP_BINS : d);
        int cls = d + RP_BINS + 1;                    // in [1, 65]
        if (!(m_i && (residue_mask[b * Lsz + j] != 0))) cls = RP_ZERO_ROW;
        offs[j] = (unsigned)cls * RP_ROW_BYTES;
    }

#if RP_HAS_TDM
    if (t < 32) __builtin_amdgcn_s_wait_tensorcnt(0); // wave0 drains its TDM op
#endif
    __syncthreads();                                  // publish T + offs to all waves

    // ---- Phase 1c: fuse bias in place: T[c][k] += bias[k]
    for (int e = t; e < RP_W_ELEMS; e += 256) {
        T[e] += bias[e & (RP_CZ - 1)];
    }
    __syncthreads();

    // ---- Phase 2: async LDS -> global streaming NT stores
    const int wave = t >> 5;                          // 8 waves per block
    const int lane = t & 31;                          // 16B chunk within a row
    float* dst_lane = out + (size_t)bi * (size_t)Lsz * RP_CZ + (size_t)lane * 4;

    for (int j0 = 0; j0 < Lsz; j0 += 8) {
        const int j = j0 + wave;                      // wave-uniform row id
        const unsigned src = tbase + offs[j] + (unsigned)(lane * 16);
        const unsigned long long dst =
            (unsigned long long)(uintptr_t)(dst_lane + (size_t)j * RP_CZ);
        // 16B per lane, 512B per wave-instruction, LDS -> memory, no VGPR data.
        asm volatile(
            "global_store_async_from_lds_b128 %0, %1, off th:TH_STORE_NT"
            :
            : "v"(dst), "v"(src)
            : "memory");
    }

    // Drain outstanding async stores before LDS can be torn down.
    asm volatile("s_wait_asynccnt 0x0" ::: "memory");
}

extern "C" void kernel_launch(void* const* d_in, const int* in_sizes, int n_in,
                              void* d_out, int out_size, void* d_ws, size_t ws_size,
                              hipStream_t stream) {
    (void)n_in; (void)d_ws; (void)ws_size; (void)out_size;
    const int*           residue_index = (const int*)d_in[0];
    const unsigned char* residue_mask  = (const unsigned char*)d_in[1];
    const float*         W             = (const float*)d_in[2];
    const float*         bias          = (const float*)d_in[3];
    float*               out           = (float*)d_out;

    const int BL = in_sizes[0];       // B * L = 1536
    const int L  = RP_L;              // 768 (reference constant)

    relpos_async_kernel<<<dim3(BL), dim3(256), 0, stream>>>(
        residue_index, residue_mask, W, bias, out, L);
}